// Conv2d_17334488006874
// MI455X (gfx1250) — compile-verified
//
#include <hip/hip_runtime.h>

typedef __attribute__((ext_vector_type(2))) float v2f;
typedef __attribute__((ext_vector_type(8))) float v8f;

#define C_IN   128
#define C_OUT  256
#define HW     64
#define SLAB_W 34                            // ow0-1 .. ow0+32 (halo)
#define SLAB_FLOATS (C_IN * 3 * SLAB_W)      // 13056 floats = 52224 B
#define BCHUNK_K 64                          // channels per weight chunk
#define BCHUNK_FLOATS (BCHUNK_K * C_OUT)     // 16384 floats = 65536 B
// A slab + double-buffered weight chunks
#define LDS_BYTES ((SLAB_FLOATS + 2 * BCHUNK_FLOATS) * 4)  // 183296 B (<320KB WGP)

typedef __attribute__((address_space(1))) int gas_int;   // global
typedef __attribute__((address_space(3))) int las_int;   // LDS

// ---- gfx1250 async global->LDS copy (per-lane LDS scatter address) ----
#if __has_builtin(__builtin_amdgcn_global_load_async_to_lds_b32)
#define GLOBAL_ASYNC_B32(SRC, DST)                                          \
  __builtin_amdgcn_global_load_async_to_lds_b32(                            \
      (gas_int*)(SRC), (las_int*)(DST), 0, 0)
#else
#define GLOBAL_ASYNC_B32(SRC, DST)                                          \
  asm volatile("global_load_async_to_lds_b32 %0, %1, off"                   \
               :: "v"((unsigned)(unsigned long long)(DST)), "v"(SRC)        \
               : "memory")
#endif

#if __has_builtin(__builtin_amdgcn_s_wait_asynccnt)
#define WAIT_ASYNC(N) __builtin_amdgcn_s_wait_asynccnt(N)
#else
#define WAIT_ASYNC(N) asm volatile("s_wait_asynccnt %0" :: "i"(N) : "memory")
#endif

// Stage one weight chunk (64 channels x 256 outs at kernel tap r) into a
// pair-swizzled LDS buffer with fire-and-forget async loads. o = tid (coalesced
// 1KB rows on the global side; 8B-stride scatter on the LDS side).
__device__ __forceinline__ void stage_b_async(const float* __restrict__ Kw,
                                              float* __restrict__ buf,
                                              int tid, int r, int c_base) {
  const int o = tid;  // blockDim == C_OUT
  const float* g0 = Kw + ((size_t)c_base * 9 + r) * C_OUT + o;
#pragma unroll 4
  for (int kk = 0; kk < BCHUNK_K; ++kk) {
    const float* src = g0 + (size_t)kk * 9 * C_OUT;           // f=(c_base+kk)*9+r
    float* dst = buf + ((((kk >> 1) * C_OUT) + o) << 1) + (kk & 1);
    GLOBAL_ASYNC_B32(src, dst);
  }
}

// Block: 256 threads = 8 waves; tile = 256 output channels x 32 pixels of one
// output row for one batch image. Wave w owns channels [w*32, w*32+32).
__global__ __launch_bounds__(256) void conv3x3_wmma_f32(
    const float* __restrict__ x,   // (32,128,64,64)
    const float* __restrict__ Kw,  // raw (128,3,3,256) == [1152][256] row-major
    float* __restrict__ out)       // (32,256,64,64)
{
  extern __shared__ float smem[];
  float* As  = smem;                      // patch slab, pair-swizzled over channel
  float* Bs0 = smem + SLAB_FLOATS;        // weight chunk buffer 0
  float* Bs1 = Bs0 + BCHUNK_FLOATS;       // weight chunk buffer 1
  const v2f* As2 = (const v2f*)As;

  const int tid   = threadIdx.x;
  const int lane  = tid & 31;
  const int lhalf = lane >> 4;
  const int lmod  = lane & 15;
  const int wv    = tid >> 5;
  const int o_base = wv * 32;

  const int owh = blockIdx.x;             // 0..1
  const int oh  = blockIdx.y;             // 0..63
  const int b   = blockIdx.z;             // 0..31
  const int ow0 = owh * 32;

  // ---- Kick off async staging of weight chunk 0 immediately ----
  stage_b_async(Kw, Bs0, tid, /*r=*/0, /*c_base=*/0);

  // ---- Stage patch slab: x[b, :, oh-1..oh+1, ow0-1..ow0+32], zero halo ----
  for (int idx = tid; idx < SLAB_FLOATS; idx += 256) {
    int j    = idx % SLAB_W;
    int rest = idx / SLAB_W;
    int dh   = rest % 3;
    int c    = rest / 3;
    int ih   = oh - 1 + dh;
    int iw   = ow0 - 1 + j;
    float v = 0.0f;
    if ((unsigned)ih < (unsigned)HW && (unsigned)iw < (unsigned)HW)
      v = x[(((size_t)b * C_IN + c) * HW + ih) * HW + iw];
    As[((((c >> 1) * 3 + dh) * SLAB_W + j) << 1) + (c & 1)] = v;
  }

  v8f acc00 = {}, acc01 = {}, acc10 = {}, acc11 = {};

  // K reordered as f' = r*128 + c : 18 chunks of 64 channels, double-buffered
  for (int q = 0; q < 18; ++q) {
    const int r      = q >> 1;
    const int c_base = (q & 1) << 6;
    const int dh = (r >= 6) ? 2 : (r >= 3 ? 1 : 0);
    const int dw = r - 3 * dh;
    float* Bcur = (q & 1) ? Bs1 : Bs0;

    // Prefetch chunk q+1 into the other buffer (safe: trailing barrier of
    // iteration q-1 guarantees every wave finished computing on it).
    if (q + 1 < 18) {
      const int rn = (q + 1) >> 1;
      const int cn = ((q + 1) & 1) << 6;
      stage_b_async(Kw, (q & 1) ? Bs0 : Bs1, tid, rn, cn);
      WAIT_ASYNC(32);   // in-order completion => all of chunk q has landed
    } else {
      WAIT_ASYNC(0);
    }
    __syncthreads();    // all waves' portions of chunk q visible (+ A slab on q==0)

    const v2f* Bs2 = (const v2f*)Bcur;
    int pIdx = ((((c_base >> 1) + lhalf) * 3 + dh) * SLAB_W) + dw + lmod;
    int wIdx = lhalf * C_OUT + o_base + lmod;
#pragma unroll
    for (int s = 0; s < 16; ++s) {        // 16 K-steps of 4 channels
      v2f bp0 = As2[pIdx];                // pixels 0..15
      v2f bp1 = As2[pIdx + 16];           // pixels 16..31
      v2f aw0 = Bs2[wIdx];                // out-channels o_base+0..15
      v2f aw1 = Bs2[wIdx + 16];           // out-channels o_base+16..31
      acc00 = __builtin_amdgcn_wmma_f32_16x16x4_f32(false, aw0, false, bp0,
                                                    (short)0, acc00, false, false);
      acc01 = __builtin_amdgcn_wmma_f32_16x16x4_f32(false, aw0, false, bp1,
                                                    (short)0, acc01, false, false);
      acc10 = __builtin_amdgcn_wmma_f32_16x16x4_f32(false, aw1, false, bp0,
                                                    (short)0, acc10, false, false);
      acc11 = __builtin_amdgcn_wmma_f32_16x16x4_f32(false, aw1, false, bp1,
                                                    (short)0, acc11, false, false);
      pIdx += 2 * 3 * SLAB_W;
      wIdx += 2 * C_OUT;
    }
    if (q + 1 < 18) __syncthreads();  // everyone done reading Bcur before the
                                      // q+2 prefetch overwrites it
  }

  // ---- Epilogue: D layout VGPR i, lane l -> o = base+i+8*lhalf, ow = +lmod ----
  v8f accArr[2][2] = {{acc00, acc01}, {acc10, acc11}};
#pragma unroll
  for (int ot = 0; ot < 2; ++ot) {
#pragma unroll
    for (int pt = 0; pt < 2; ++pt) {
      int o0 = o_base + ot * 16 + 8 * lhalf;
      int ow = ow0 + pt * 16 + lmod;
      bool zero = (oh == HW - 1) || (ow == HW - 1);  // reference zeroes last row/col
      size_t base = (((size_t)b * C_OUT + o0) * HW + oh) * HW + ow;
#pragma unroll
      for (int i = 0; i < 8; ++i) {
        float v = zero ? 0.0f : accArr[ot][pt][i];
        out[base + (size_t)i * HW * HW] = v;
      }
    }
  }
}

extern "C" void kernel_launch(void* const* d_in, const int* in_sizes, int n_in,
                              void* d_out, int out_size, void* d_ws, size_t ws_size,
                              hipStream_t stream) {
  const float* x  = (const float*)d_in[0];
  const float* Kw = (const float*)d_in[1];
  float* out = (float*)d_out;

  (void)hipFuncSetAttribute((const void*)conv3x3_wmma_f32,
                            hipFuncAttributeMaxDynamicSharedMemorySize, LDS_BYTES);

  dim3 grid(2, HW, 32);   // (half-row, oh, batch) = 4096 blocks
  conv3x3_wmma_f32<<<grid, 256, LDS_BYTES, stream>>>(x, Kw, out);
}